// TurboQuantRotation_47837345743070
// MI455X (gfx1250) — compile-verified
//
#include <hip/hip_runtime.h>

// Cascaded SRHT for MI455X (gfx1250).
//  * H_2048 = H_16 (x) H_16 (x) H_8 (Sylvester Kronecker factorization) ->
//    two exact fp32 16x16 matmuls (V_WMMA_F32_16X16X4_F32 chains) + H8
//    butterflies per token. ~0.74 GFLOP total vs 275 GFLOP dense.
//  * HBM-bound: 268 MB traffic -> ~11.5us floor @ 23.3 TB/s. One wave per
//    token; 8KB token tile lives in LDS for both passes.
//  * HBM<->LDS moved with gfx1250 async ops (GLOBAL_LOAD_ASYNC_TO_LDS_B128 /
//    GLOBAL_STORE_ASYNC_FROM_LDS_B128, ASYNCcnt) -- no VGPR round-trip.
//  * Sign flips are sign-bit XORs in LDS; final 1/2048 scale folded into the
//    last butterfly stage; wht_mat input never read.

typedef __attribute__((ext_vector_type(2))) float v2f;
typedef __attribute__((ext_vector_type(8))) float v8f;

#define SRHT_D 2048
#define SRHT_TOKENS 16384
#define WAVES_PER_BLOCK 8
#define BLOCK_THREADS (WAVES_PER_BLOCK * 32)

// +1.0f / -1.0f from Sylvester parity, branchless (no v_cmp/v_cndmask).
__device__ __forceinline__ float hsgnf(int i, int j) {
  unsigned parity = (unsigned)(__builtin_popcount(i & j) & 1);
  return __uint_as_float(0x3F800000u | (parity << 31));
}

__global__ __launch_bounds__(BLOCK_THREADS)
void srht_wmma_kernel(const float* __restrict__ x,
                      const int* __restrict__ signs,
                      float* __restrict__ out) {
  __shared__ float lds[WAVES_PER_BLOCK * SRHT_D];   // 64 KB

  const int lane  = threadIdx.x & 31;
  const int wave  = threadIdx.x >> 5;
  const int token = blockIdx.x * WAVES_PER_BLOCK + wave;
  float* t = lds + wave * SRHT_D;

  const int hi16 = lane >> 4;   // lane half
  const int nn   = lane & 15;   // M for A-frag, N for B/C/D frags

  // H16 A-fragments, 4 K-chunks of 4 (ISA 16x4 f32 A layout:
  // lane L holds M=L%16, K = 4k + 2*(L/16) + {0,1}).
  v2f afrag[4];
#pragma unroll
  for (int k = 0; k < 4; ++k) {
    const int K0 = 4 * k + 2 * hi16;
    afrag[k].x = hsgnf(nn, K0);
    afrag[k].y = hsgnf(nn, K0 + 1);
  }

  // LDS byte offset of this wave's tile (generic LDS pointer: low 32 bits
  // are the in-allocation byte offset per the flat-address LDS aperture).
  const unsigned lds_base = (unsigned)(uintptr_t)t + (unsigned)(lane * 16);

  // ---- Async stream HBM -> LDS: 16 x GLOBAL_LOAD_ASYNC_TO_LDS_B128 ----
  {
    const float* g = x + (size_t)token * SRHT_D + lane * 4;
    unsigned la = lds_base;
#pragma unroll
    for (int i = 0; i < 16; ++i) {
      asm volatile("global_load_async_to_lds_b128 %0, %1, off"
                   :: "v"(la), "v"(g)
                   : "memory");
      la += 512;
      g += 128;
    }
    asm volatile("s_wait_asynccnt 0x0" ::: "memory");
  }
  __syncthreads();

#pragma unroll
  for (int p = 0; p < 2; ++p) {
    // ---- Sign flip in LDS: xor sign bit (signs are +/-1 int32) ----
    const int* sp = signs + p * SRHT_D;
#pragma unroll
    for (int i = 0; i < 16; ++i) {
      const int d0 = i * 128 + lane * 4;
      int4 sv = *(const int4*)(sp + d0);
      int4 v  = *(const int4*)(t + d0);
      v.x ^= (sv.x & (int)0x80000000);
      v.y ^= (sv.y & (int)0x80000000);
      v.z ^= (sv.z & (int)0x80000000);
      v.w ^= (sv.w & (int)0x80000000);
      *(int4*)(t + d0) = v;
    }
    __syncthreads();

    // ---- Stage A: contract 'a' with H16   (d = a*128 + l) ----
#pragma unroll 2
    for (int n = 0; n < 8; ++n) {
      v8f acc = {0.f, 0.f, 0.f, 0.f, 0.f, 0.f, 0.f, 0.f};
#pragma unroll
      for (int k = 0; k < 4; ++k) {
        const int K0 = 4 * k + 2 * hi16;
        v2f bfrag;
        bfrag.x = t[K0 * 128 + n * 16 + nn];
        bfrag.y = t[(K0 + 1) * 128 + n * 16 + nn];
        acc = __builtin_amdgcn_wmma_f32_16x16x4_f32(
            false, afrag[k], false, bfrag, (short)0, acc, false, false);
      }
#pragma unroll
      for (int r = 0; r < 8; ++r) {
        const int M = r + 8 * hi16;
        t[M * 128 + n * 16 + nn] = acc[r];
      }
    }
    __syncthreads();

    // ---- Stage B: contract 'b' with H16   (d = a*128 + b*8 + c) ----
#pragma unroll 2
    for (int n = 0; n < 8; ++n) {
      const int m  = n * 16 + nn;
      const int ap = m >> 3;
      const int cc = m & 7;
      v8f acc = {0.f, 0.f, 0.f, 0.f, 0.f, 0.f, 0.f, 0.f};
#pragma unroll
      for (int k = 0; k < 4; ++k) {
        const int K0 = 4 * k + 2 * hi16;
        v2f bfrag;
        bfrag.x = t[ap * 128 + K0 * 8 + cc];
        bfrag.y = t[ap * 128 + (K0 + 1) * 8 + cc];
        acc = __builtin_amdgcn_wmma_f32_16x16x4_f32(
            false, afrag[k], false, bfrag, (short)0, acc, false, false);
      }
#pragma unroll
      for (int r = 0; r < 8; ++r) {
        const int M = r + 8 * hi16;
        t[ap * 128 + M * 8 + cc] = acc[r];
      }
    }
    __syncthreads();

    // ---- Stage C: contract 'c' with H8 (butterflies); fold final scale ----
    const float cs = (p == 1) ? (1.0f / 2048.0f) : 1.0f;
#pragma unroll
    for (int g = 0; g < 8; ++g) {
      const int base = (g * 32 + lane) * 8;  // lane-exclusive group of 8
      float v0 = t[base + 0], v1 = t[base + 1], v2 = t[base + 2],
            v3 = t[base + 3], v4 = t[base + 4], v5 = t[base + 5],
            v6 = t[base + 6], v7 = t[base + 7];
      float b0 = v0 + v1, b1 = v0 - v1, b2 = v2 + v3, b3 = v2 - v3;
      float b4 = v4 + v5, b5 = v4 - v5, b6 = v6 + v7, b7 = v6 - v7;
      float c0 = b0 + b2, c1 = b1 + b3, c2 = b0 - b2, c3 = b1 - b3;
      float c4 = b4 + b6, c5 = b5 + b7, c6 = b4 - b6, c7 = b5 - b7;
      t[base + 0] = (c0 + c4) * cs;
      t[base + 1] = (c1 + c5) * cs;
      t[base + 2] = (c2 + c6) * cs;
      t[base + 3] = (c3 + c7) * cs;
      t[base + 4] = (c0 - c4) * cs;
      t[base + 5] = (c1 - c5) * cs;
      t[base + 6] = (c2 - c6) * cs;
      t[base + 7] = (c3 - c7) * cs;
    }
    __syncthreads();
  }

  // ---- Async stream LDS -> HBM: 16 x GLOBAL_STORE_ASYNC_FROM_LDS_B128 ----
  {
    float* go = out + (size_t)token * SRHT_D + lane * 4;
    unsigned la = lds_base;
    asm volatile("s_wait_dscnt 0x0" ::: "memory");  // LDS writes visible
#pragma unroll
    for (int i = 0; i < 16; ++i) {
      asm volatile("global_store_async_from_lds_b128 %0, %1, off"
                   :: "v"(go), "v"(la)
                   : "memory");
      la += 512;
      go += 128;
    }
    asm volatile("s_wait_asynccnt 0x0" ::: "memory");
  }
}

extern "C" void kernel_launch(void* const* d_in, const int* in_sizes, int n_in,
                              void* d_out, int out_size, void* d_ws, size_t ws_size,
                              hipStream_t stream) {
  const float* x   = (const float*)d_in[0];  // [16384, 2048] fp32
  const int* signs = (const int*)d_in[1];    // [2, 2048] int32 (+/-1)
  // d_in[2] = wht_mat: intentionally unused (Kronecker-factored in-kernel)
  float* out = (float*)d_out;                // [16384, 2048] fp32

  dim3 grid(SRHT_TOKENS / WAVES_PER_BLOCK);  // 2048 blocks, exact
  srht_wmma_kernel<<<grid, BLOCK_THREADS, 0, stream>>>(x, signs, out);
}